// GCNDecoder_25632364822536
// MI455X (gfx1250) — compile-verified
//
#include <hip/hip_runtime.h>

typedef float v2f __attribute__((ext_vector_type(2)));
typedef float v8f __attribute__((ext_vector_type(8)));

// ---------------------------------------------------------------- elementwise
__global__ __launch_bounds__(256) void k_fill(float* __restrict__ p, float v, int n) {
  int i = blockIdx.x * 256 + threadIdx.x;
  if (i < n) p[i] = v;
}

__global__ __launch_bounds__(256) void k_deg(const int* __restrict__ col,
                                             const float* __restrict__ ew,
                                             float* __restrict__ deg, int E) {
  int e = blockIdx.x * 256 + threadIdx.x;
  if (e < E) atomicAdd(&deg[col[e]], ew[e]);
}

__global__ __launch_bounds__(256) void k_dinv(float* __restrict__ deg, int n) {
  int i = blockIdx.x * 256 + threadIdx.x;
  if (i < n) {
    float d = deg[i];
    deg[i] = (d > 0.f) ? rsqrtf(d) : 0.f;
  }
}

// agg[i][:] = t[i][:] * dinv[i]^2   (self-loop contribution, norm = dinv*1*dinv)
__global__ __launch_bounds__(256) void k_init_agg(const float* __restrict__ t,
                                                  const float* __restrict__ dinv,
                                                  float* __restrict__ agg, long n64) {
  long i = (long)blockIdx.x * 256 + threadIdx.x;
  if (i < n64) {
    float s = dinv[i >> 6];
    agg[i] = t[i] * s * s;
  }
}

__global__ __launch_bounds__(256) void k_relu(float* __restrict__ p, long n) {
  long i = (long)blockIdx.x * 256 + threadIdx.x;
  if (i < n) p[i] = fmaxf(p[i], 0.f);
}

// one wave (32 lanes) per edge; each lane handles 2 of the 64 features
__global__ __launch_bounds__(256) void k_scatter(const float* __restrict__ h,
                                                 const int* __restrict__ row,
                                                 const int* __restrict__ col,
                                                 const float* __restrict__ ew,
                                                 const float* __restrict__ dinv,
                                                 float* __restrict__ agg, int E) {
  int lane = threadIdx.x & 31;
  int e = blockIdx.x * 8 + (threadIdx.x >> 5);
  if (e >= E) return;
  e = __builtin_amdgcn_readfirstlane(e);   // wave-uniform -> scalar loads
  int r = row[e];
  int c = col[e];
  float nrm = dinv[r] * ew[e] * dinv[c];
  v2f v = *(const v2f*)(h + (size_t)r * 64 + lane * 2);
  float* dst = agg + (size_t)c * 64 + lane * 2;
  atomicAdd(dst + 0, v.x * nrm);
  atomicAdd(dst + 1, v.y * nrm);
}

// ---------------------------------------------------------------- WMMA GEMM
// D[M x NCOLS] = A[M x 64] @ B[64 x NCOLS] (+ bias, + relu).  fp32 WMMA.
// One wave computes one 16x16 output tile; K-loop of 16 x (16x16x4) WMMAs.
template <int NCOLS, bool BIAS, bool RELU>
__global__ __launch_bounds__(256) void k_gemm(const float* __restrict__ A,
                                              const float* __restrict__ B,
                                              const float* __restrict__ bias,
                                              float* __restrict__ D, int M) {
  const int lane = threadIdx.x & 31;
  const int tile = blockIdx.x * 8 + (threadIdx.x >> 5);
  constexpr int NT = NCOLS / 16;
  const int mt = tile / NT;
  const int nt = tile % NT;
  if (mt * 16 >= M) return;                // wave-uniform: EXEC stays all-1s
  const int tm = mt * 16, tn = nt * 16;
  const int half = lane >> 4;              // 0: lanes 0-15, 1: lanes 16-31
  const int lr = lane & 15;

  int arow = tm + lr;
  if (arow >= M) arow = M - 1;             // clamp (harmless duplicate read)
  const float* ap = A + (size_t)arow * 64;
  const float* bp = B + tn + lr;

  v8f acc = {};
#pragma unroll
  for (int k0 = 0; k0 < 64; k0 += 4) {
    const int ka = k0 + 2 * half;          // lanes 0-15: K={k0,k0+1}; 16-31: K={k0+2,k0+3}
    v2f a = *(const v2f*)(ap + ka);        // A fragment (16x4 f32, 2 VGPRs)
    v2f b;                                 // B fragment (4x16 f32, 2 VGPRs)
    b.x = bp[(size_t)(ka + 0) * NCOLS];
    b.y = bp[(size_t)(ka + 1) * NCOLS];
    acc = __builtin_amdgcn_wmma_f32_16x16x4_f32(false, a, false, b,
                                                (short)0, acc, false, false);
  }

  float bv = 0.f;
  if (BIAS) bv = bias[tn + lr];
#pragma unroll
  for (int rr = 0; rr < 8; ++rr) {         // C/D: VGPR rr -> M = rr + 8*half
    int orow = tm + rr + half * 8;
    if (orow < M) {
      float v = acc[rr] + bv;
      if (RELU) v = fmaxf(v, 0.f);
      D[(size_t)orow * NCOLS + tn + lr] = v;
    }
  }
}

// ---------------------------------------------------------------- launch
extern "C" void kernel_launch(void* const* d_in, const int* in_sizes, int n_in,
                              void* d_out, int out_size, void* d_ws, size_t ws_size,
                              hipStream_t stream) {
  (void)n_in; (void)out_size; (void)ws_size;
  const float* x  = (const float*)d_in[0];
  const int*   ei = (const int*)d_in[1];
  const float* ew = (const float*)d_in[2];
  const float* W1 = (const float*)d_in[3];
  const float* W2 = (const float*)d_in[4];
  const float* A1 = (const float*)d_in[5];
  const float* b1 = (const float*)d_in[6];
  const float* A2 = (const float*)d_in[7];
  const float* b2 = (const float*)d_in[8];

  const int N = in_sizes[0] / 64;          // 100000
  const int E = in_sizes[2];               // 1600000
  const int* row = ei;                     // edge_index[0]
  const int* col = ei + E;                 // edge_index[1]

  float* r_out = (float*)d_out;            // [N, 12*32] flattened
  float* h_out = r_out + (size_t)N * 384;  // [N, 64]

  // workspace: dinv (N) | bufA (N*64) | bufB (N*64)  ~= 51.6 MB
  char* ws = (char*)d_ws;
  size_t off = 0;
  auto alloc = [&](size_t bytes) {
    void* p = ws + off;
    off = (off + bytes + 255) & ~(size_t)255;
    return p;
  };
  float* dinv = (float*)alloc((size_t)N * 4);
  float* bufA = (float*)alloc((size_t)N * 64 * 4);
  float* bufB = (float*)alloc((size_t)N * 64 * 4);

  const long n64 = (long)N * 64;
  dim3 blk(256);
  const int gN    = (N + 255) / 256;
  const int gE    = (E + 255) / 256;
  const int gN64  = (int)((n64 + 255) / 256);
  const int gSc   = (E + 7) / 8;                 // 8 waves / block, 1 edge / wave
  const int mt    = (N + 15) / 16;
  const int gG64  = (mt * 4  + 7) / 8;           // 64-col GEMM tiles
  const int gG384 = (mt * 24 + 7) / 8;           // 384-col GEMM tiles

  // degree -> dinv  (self-loop contributes weight 1.0)
  k_fill<<<gN, blk, 0, stream>>>(dinv, 1.0f, N);
  k_deg <<<gE, blk, 0, stream>>>(col, ew, dinv, E);
  k_dinv<<<gN, blk, 0, stream>>>(dinv, N);

  // GCN layer 1: bufA = x@W1 ; bufB = aggregate ; relu -> H1
  k_gemm<64, false, false><<<gG64, blk, 0, stream>>>(x, W1, nullptr, bufA, N);
  k_init_agg<<<gN64, blk, 0, stream>>>(bufA, dinv, bufB, n64);
  k_scatter <<<gSc,  blk, 0, stream>>>(bufA, row, col, ew, dinv, bufB, E);
  k_relu    <<<gN64, blk, 0, stream>>>(bufB, n64);

  // GCN layer 2: bufA = H1@W2 ; accumulate directly into h-output region
  k_gemm<64, false, false><<<gG64, blk, 0, stream>>>(bufB, W2, nullptr, bufA, N);
  k_init_agg<<<gN64, blk, 0, stream>>>(bufA, dinv, h_out, n64);
  k_scatter <<<gSc,  blk, 0, stream>>>(bufA, row, col, ew, dinv, h_out, E);
  k_relu    <<<gN64, blk, 0, stream>>>(h_out, n64);        // h_out = H2 (output h)

  // readout: bufB = relu(H2@A1 + b1) ; r_out = bufB@A2 + b2
  k_gemm<64,  true, true ><<<gG64,  blk, 0, stream>>>(h_out, A1, b1, bufB, N);
  k_gemm<384, true, false><<<gG384, blk, 0, stream>>>(bufB, A2, b2, r_out, N);
}